// EmbeddingLSQ_57818849739271
// MI455X (gfx1250) — compile-verified
//
#include <hip/hip_runtime.h>

// EmbeddingLSQ on MI455X (gfx1250, wave32).
//
// out[t, d] = (idx[t]==0) ? 0 : round(clamp(W[d, idx[t]]/alpha, -8, 7)) * alpha
// where idx[t] = argmax_v x[t, v] and x is one-hot.
//
// Memory-bound gather. Traffic budget @ 23.3 TB/s:
//   x scan:  512MB worst case, ~288MB expected with chunked early-exit (NT loads)
//   weight:  <=128MB, kept L2-resident (192MB L2) via regular-temporal hints
//   out:     16MB, NT stores (never re-read)
// => ~18.5us expected roofline. The one-hot einsum reduces algebraically to a
// gather, so WMMA is inapplicable; the CDNA5 levers are coalesced B128 NT
// streaming, temporal-hint cache partitioning, and early-exit to halve x reads.

#define VOCAB 32000
#define DIM   1024
#define QN   -8.0f
#define QP    7.0f

typedef unsigned int uintv4 __attribute__((ext_vector_type(4)));

__global__ __launch_bounds__(256) void embedding_lsq_kernel(
    const float* __restrict__ x,       // [T, VOCAB] one-hot f32
    const float* __restrict__ weight,  // [DIM, VOCAB] f32
    const float* __restrict__ alpha,   // [1] f32
    float* __restrict__ out)           // [T, DIM] f32
{
    const int t = blockIdx.x;
    const int tid = threadIdx.x;

    __shared__ int s_idx;
    if (tid == 0) s_idx = -1;  // -1 = not found (0 is a valid index: PAD)
    __syncthreads();

    // ---- Phase 1: find the one-hot index for token t, with early exit. ----
    // Row = 32000 f32 = 128000 B = 8000 uint4, 16B-aligned for every t.
    // 7 full chunks of 1024 uint4 (16KB: 256 thr x 4 NT global_load_b128 in
    // flight) + 832-vector remainder. After each chunk, the block uniformly
    // checks the found flag between two barriers (write -> bar -> read -> bar
    // -> uniform break), so no wave can race a new write against the read.
    // 1.0f = 0x3F800000, 0.0f = 0 -> integer OR test is exact.
    const uintv4* __restrict__ xrow =
        reinterpret_cast<const uintv4*>(x + (size_t)t * VOCAB);

    int found = -1;
    for (int chunk = 0; chunk < 7; ++chunk) {
        const int base = chunk * 1024 + tid;
        uintv4 u0 = __builtin_nontemporal_load(xrow + base);
        uintv4 u1 = __builtin_nontemporal_load(xrow + base + 256);
        uintv4 u2 = __builtin_nontemporal_load(xrow + base + 512);
        uintv4 u3 = __builtin_nontemporal_load(xrow + base + 768);
        if (u0.x | u0.y | u0.z | u0.w) {
            int c = base * 4;
            s_idx = u0.x ? c : (u0.y ? (c + 1) : (u0.z ? (c + 2) : (c + 3)));
        }
        if (u1.x | u1.y | u1.z | u1.w) {
            int c = (base + 256) * 4;
            s_idx = u1.x ? c : (u1.y ? (c + 1) : (u1.z ? (c + 2) : (c + 3)));
        }
        if (u2.x | u2.y | u2.z | u2.w) {
            int c = (base + 512) * 4;
            s_idx = u2.x ? c : (u2.y ? (c + 1) : (u2.z ? (c + 2) : (c + 3)));
        }
        if (u3.x | u3.y | u3.z | u3.w) {
            int c = (base + 768) * 4;
            s_idx = u3.x ? c : (u3.y ? (c + 1) : (u3.z ? (c + 2) : (c + 3)));
        }
        __syncthreads();        // all writes of this chunk visible
        found = s_idx;          // every thread reads the same value
        __syncthreads();        // no next-chunk write until all have read
        if (found >= 0) break;  // uniform across the block
    }
    if (found < 0) {
        // Remainder: uint4 indices [7168, 8000). Only reached block-uniformly.
        for (int i = 7168 + tid; i < VOCAB / 4; i += 256) {
            uintv4 u = __builtin_nontemporal_load(xrow + i);
            if (u.x | u.y | u.z | u.w) {
                int c = i * 4;
                s_idx = u.x ? c : (u.y ? (c + 1) : (u.z ? (c + 2) : (c + 3)));
            }
        }
        __syncthreads();
        found = s_idx;
    }

    const int v = (found < 0) ? 0 : found;  // one-hot guarantees found >= 0
    float* __restrict__ orow = out + (size_t)t * DIM;

    // ---- Phase 2: gather column v of W, LSQ fake-quantize, store. ----
    if (v == 0) {
        // PAD token: masked to zero. NT stores (out never re-read).
        #pragma unroll
        for (int k = 0; k < DIM / 256; ++k) {
            __builtin_nontemporal_store(0.0f, &orow[tid + k * 256]);
        }
    } else {
        const float a = alpha[0];  // uniform -> scalar load
        // Regular-temporal gathers: weight (128MB) fits in the 192MB L2 and
        // is re-gathered ~4M times -- keep it cached.
        #pragma unroll
        for (int k = 0; k < DIM / 256; ++k) {
            int d = tid + k * 256;
            float wv = weight[(size_t)d * VOCAB + v];  // 128KB-strided, L2-resident
            float q = rintf(fminf(fmaxf(wv / a, QN), QP));  // RNE, matches jnp.round
            __builtin_nontemporal_store(q * a, &orow[d]);
        }
    }
}

extern "C" void kernel_launch(void* const* d_in, const int* in_sizes, int n_in,
                              void* d_out, int out_size, void* d_ws, size_t ws_size,
                              hipStream_t stream) {
    const float* x      = (const float*)d_in[0];  // [T, VOCAB]
    const float* weight = (const float*)d_in[1];  // [DIM, VOCAB]
    const float* alpha  = (const float*)d_in[2];  // [1]
    float* out = (float*)d_out;                   // [T, DIM]

    const int T = in_sizes[0] / VOCAB;  // 4096

    // One workgroup (8 wave32s) per token: phase-1 scan is fully coalesced
    // with early exit, 4096 concurrent blocks hide column-gather latency.
    embedding_lsq_kernel<<<T, 256, 0, stream>>>(x, weight, alpha, out);
}